// GatingNetwork_46617575031149
// MI455X (gfx1250) — compile-verified
//
#include <hip/hip_runtime.h>
#include <math.h>

typedef __attribute__((ext_vector_type(2))) float v2f;
typedef __attribute__((ext_vector_type(4))) float v4f;
typedef __attribute__((ext_vector_type(8))) float v8f;

#define D_MODEL     2048
#define NUM_EXPERTS 64
#define N_TOKENS    16384
#define KC          128           // K-chunk staged in LDS
#define KP          (KC / 2)      // k-pairs per chunk
#define WAVES       4
#define BLOCK_T     (WAVES * 32)  // 128 threads
#define ROWS_BLOCK  (WAVES * 16)  // 64 tokens per block

__device__ __forceinline__ void merge_top2(float& v0, int& i0, float& v1, int& i1,
                                           float u, int j) {
  if (u > v0 || (u == v0 && j < i0)) { v1 = v0; i1 = i0; v0 = u; i0 = j; }
  else if (u > v1 || (u == v1 && j < i1)) { v1 = u; i1 = j; }
}

__global__ __launch_bounds__(BLOCK_T) void gate_topk_kernel(
    const float* __restrict__ x, const float* __restrict__ W,
    const float* __restrict__ bias, float* __restrict__ out) {
  // W chunk staged as k-pairs: wlds[kp*64 + c] = (W[k0+2kp][c], W[k0+2kp+1][c])
  __shared__ v2f wlds[KP * NUM_EXPERTS];  // 32 KB

  const int tid  = threadIdx.x;
  const int wave = tid >> 5;
  const int lane = tid & 31;
  const int col  = lane & 15;   // column within 16-wide N tile / A row
  const int h    = lane >> 4;   // lane half: selects K-pair {0,1} vs {2,3}

  const int base_m = blockIdx.x * ROWS_BLOCK + wave * 16;
  const float* xrow = x + (size_t)(base_m + col) * D_MODEL;

  v8f acc[4];
  const v8f zero8 = {};
#pragma unroll
  for (int t = 0; t < 4; ++t) acc[t] = zero8;

  // Staging decomposition: 16 col-groups of 4, BLOCK_T/16 kp-rows per pass
  const int cg  = (tid & 15) * 4;
  const int kp0 = tid >> 4;

  for (int k0 = 0; k0 < D_MODEL; k0 += KC) {
    __syncthreads();  // previous chunk fully consumed
#pragma unroll 2
    for (int kp = kp0; kp < KP; kp += (BLOCK_T >> 4)) {
      const float* wr0 = W + (size_t)(k0 + 2 * kp) * NUM_EXPERTS + cg;
      v4f wa = *(const v4f*)wr0;                 // W[2kp  ][cg..cg+3]
      v4f wb = *(const v4f*)(wr0 + NUM_EXPERTS); // W[2kp+1][cg..cg+3]
      v4f p0 = {wa.x, wb.x, wa.y, wb.y};
      v4f p1 = {wa.z, wb.z, wa.w, wb.w};
      v4f* dst = (v4f*)&wlds[kp * NUM_EXPERTS + cg];
      dst[0] = p0;
      dst[1] = p1;
    }
    __syncthreads();

    if (k0 + KC < D_MODEL)
      __builtin_prefetch(xrow + k0 + KC, 0, 3);  // global_prefetch_b8 next x chunk

#pragma unroll 4
    for (int ks = 0; ks < KC; ks += 4) {
      // A tile: x[base_m+col][k0+ks+2h .. +1]  (16x4 fp32 layout)
      v2f a = *(const v2f*)(xrow + k0 + ks + 2 * h);
      const v2f* bp = &wlds[((ks >> 1) + h) * NUM_EXPERTS + col];
#pragma unroll
      for (int t = 0; t < 4; ++t) {
        v2f bm = bp[t * 16];  // (W[kk][16t+col], W[kk+1][16t+col])
        acc[t] = __builtin_amdgcn_wmma_f32_16x16x4_f32(
            false, a, false, bm, (short)0, acc[t], false, false);
      }
    }
  }

  // bias per column (same for both lane halves)
  float bv[4];
#pragma unroll
  for (int t = 0; t < 4; ++t) bv[t] = bias[t * 16 + col];

  // outputs concatenated flat: topk_weights [T,2], topk_indices [T,2], weights [T,64]
  float* out_tw = out;
  float* out_ti = out + 2 * (size_t)N_TOKENS;
  float* out_p  = out + 4 * (size_t)N_TOKENS;

#pragma unroll
  for (int r = 0; r < 8; ++r) {
    // C/D layout: reg r holds row r (lanes 0-15) and row r+8 (lanes 16-31)
    const int tok = base_m + r + 8 * h;
    float l[4];
#pragma unroll
    for (int t = 0; t < 4; ++t) l[t] = acc[t][r] + bv[t];

    // row max across 4 local values and the 16 lanes of this half
    float m = fmaxf(fmaxf(l[0], l[1]), fmaxf(l[2], l[3]));
#pragma unroll
    for (int mask = 1; mask <= 8; mask <<= 1)
      m = fmaxf(m, __shfl_xor(m, mask, 32));

    float e[4], s = 0.f;
#pragma unroll
    for (int t = 0; t < 4; ++t) { e[t] = __expf(l[t] - m); s += e[t]; }
#pragma unroll
    for (int mask = 1; mask <= 8; mask <<= 1)
      s += __shfl_xor(s, mask, 32);
    const float inv = 1.f / s;

    // full softmax weights + local top-2 among this lane's 4 columns
    float v0 = -1.f, v1 = -1.f;
    int   i0 = 0,    i1 = 0;
#pragma unroll
    for (int t = 0; t < 4; ++t) {
      const float wgt = e[t] * inv;
      out_p[(size_t)tok * NUM_EXPERTS + t * 16 + col] = wgt;
      const int id = t * 16 + col;
      if (wgt > v0)      { v1 = v0; i1 = i0; v0 = wgt; i0 = id; }
      else if (wgt > v1) { v1 = wgt; i1 = id; }
    }

    // butterfly merge of top-2 across the 16 lanes of this half
#pragma unroll
    for (int mask = 1; mask <= 8; mask <<= 1) {
      float u0 = __shfl_xor(v0, mask, 32);
      float u1 = __shfl_xor(v1, mask, 32);
      int   j0 = __shfl_xor(i0, mask, 32);
      int   j1 = __shfl_xor(i1, mask, 32);
      merge_top2(v0, i0, v1, i1, u0, j0);
      merge_top2(v0, i0, v1, i1, u1, j1);
    }

    if (col == 0) {
      out_tw[(size_t)tok * 2 + 0] = v0;
      out_tw[(size_t)tok * 2 + 1] = v1;
      // int32 indices promote to fp32 in the concatenated flat output
      out_ti[(size_t)tok * 2 + 0] = (float)i0;
      out_ti[(size_t)tok * 2 + 1] = (float)i1;
    }
  }
}

extern "C" void kernel_launch(void* const* d_in, const int* in_sizes, int n_in,
                              void* d_out, int out_size, void* d_ws, size_t ws_size,
                              hipStream_t stream) {
  (void)in_sizes; (void)n_in; (void)d_ws; (void)ws_size; (void)out_size;
  const float* x    = (const float*)d_in[0];
  const float* W    = (const float*)d_in[1];
  const float* bias = (const float*)d_in[2];
  float* out        = (float*)d_out;

  dim3 grid(N_TOKENS / ROWS_BLOCK);  // 256 blocks
  dim3 block(BLOCK_T);               // 128 threads = 4 waves
  gate_topk_kernel<<<grid, block, 0, stream>>>(x, W, bias, out);
}